// GATModel_85169201480432
// MI455X (gfx1250) — compile-verified
//
#include <hip/hip_runtime.h>

#define NNODES 100000
#define NEDGES 1600000
#define FIN    16
#define NH     8
#define NC     8
#define HC     64                      // NH * NC
#define ETOT   (NEDGES + NNODES)       // edges + self loops

typedef __attribute__((ext_vector_type(2))) float v2f;
typedef __attribute__((ext_vector_type(8))) float v8f;

// ---------- ordered-uint encoding so unsigned atomicMax == float max ----------
__device__ __forceinline__ unsigned f2ord(float f) {
    unsigned b = __float_as_uint(f);
    return (b & 0x80000000u) ? ~b : (b | 0x80000000u);
}
__device__ __forceinline__ float ord2f(unsigned k) {
    unsigned b = (k & 0x80000000u) ? (k ^ 0x80000000u) : ~k;
    return __uint_as_float(b);
}

// ---------- init: zero accumulators (harness poisons buffers once; we must
// ---------- re-zero every call so graph replay is deterministic) ----------
__global__ void k_init(float* __restrict__ out, float* __restrict__ denom,
                       unsigned* __restrict__ emax) {
    int i = blockIdx.x * blockDim.x + threadIdx.x;
    if (i < NNODES * HC) out[i] = 0.0f;
    if (i < NNODES * NH) { denom[i] = 0.0f; emax[i] = 0u; }   // key 0 == -inf-ish
}

// ---------- h = x @ W via V_WMMA_F32_16X16X4_F32 (full fp32) ----------
// One wave handles a 16-row tile of x against all 64 output columns
// (4 column tiles, K=16 chained as 4 k-steps of 4).
__global__ __launch_bounds__(256) void k_gemm(const float* __restrict__ x,
                                              const float* __restrict__ W,
                                              float* __restrict__ h) {
    const int lane = threadIdx.x & 31;
    const int wave = threadIdx.x >> 5;
    const int tile = blockIdx.x * 8 + wave;
    const long base = (long)tile * 16;
    if (base >= NNODES) return;                 // uniform per wave: EXEC stays all-1

    const int r  = lane & 15;                   // row (A) / col (B,D) within tile
    const int kh = (lane >> 4) << 1;            // lanes 0-15: K base 0; 16-31: K base 2

    // A fragments (16x4 fp32): lane<16 holds K=0,1 ; lane>=16 holds K=2,3
    const float* xr = x + (base + r) * FIN;
    v2f a[4];
#pragma unroll
    for (int s = 0; s < 4; ++s) {
        const int k0 = s * 4 + kh;
        a[s][0] = xr[k0];
        a[s][1] = xr[k0 + 1];
    }

#pragma unroll
    for (int j = 0; j < 4; ++j) {               // 4 column tiles of 16
        v8f acc = {};
#pragma unroll
        for (int s = 0; s < 4; ++s) {           // chain K: 4 x (k=4)
            const int k0 = s * 4 + kh;
            v2f b;
            b[0] = W[(k0    ) * HC + j * 16 + r];
            b[1] = W[(k0 + 1) * HC + j * 16 + r];
            acc = __builtin_amdgcn_wmma_f32_16x16x4_f32(
                /*neg_a=*/false, a[s], /*neg_b=*/false, b,
                /*c_mod=*/(short)0, acc, /*reuse_a=*/false, /*reuse_b=*/false);
        }
        // D layout: lanes 0-15 -> M = v ; lanes 16-31 -> M = v + 8 ; N = r
#pragma unroll
        for (int v = 0; v < 8; ++v) {
            const int row = ((lane >> 4) << 3) + v;
            h[(base + row) * HC + j * 16 + r] = acc[v];
        }
    }
}

// ---------- per-node attention coefficients ----------
__global__ __launch_bounds__(256) void k_att(const float* __restrict__ h,
                                             const float* __restrict__ att_src,
                                             const float* __restrict__ att_dst,
                                             float* __restrict__ a_src,
                                             float* __restrict__ a_dst) {
    int i = blockIdx.x * blockDim.x + threadIdx.x;   // over N*H
    if (i >= NNODES * NH) return;
    const int hh = i & 7;
    const float* hp = h + (long)(i >> 3) * HC + hh * NC;
    float ss = 0.0f, sd = 0.0f;
#pragma unroll
    for (int c = 0; c < NC; ++c) {
        float v = hp[c];
        ss = fmaf(v, att_src[hh * NC + c], ss);
        sd = fmaf(v, att_dst[hh * NC + c], sd);
    }
    a_src[i] = ss;
    a_dst[i] = sd;
}

// ---------- edge helpers ----------
__device__ __forceinline__ void edge_sd(const int* __restrict__ ei, int e,
                                        int& s, int& d) {
    if (e < NEDGES) { s = ei[e]; d = ei[NEDGES + e]; }
    else            { s = d = e - NEDGES; }            // self loop
}
__device__ __forceinline__ float edge_e(const float* __restrict__ as,
                                        const float* __restrict__ ad,
                                        int s, int d, int hh) {
    float v = as[s * NH + hh] + ad[d * NH + hh];
    return v > 0.0f ? v : 0.2f * v;                    // leaky_relu(0.2)
}

// ---------- pass 1: segment max (ordered-uint atomic max) ----------
__global__ __launch_bounds__(256) void k_emax(const int* __restrict__ ei,
                                              const float* __restrict__ as,
                                              const float* __restrict__ ad,
                                              unsigned* __restrict__ emax) {
    int i = blockIdx.x * blockDim.x + threadIdx.x;     // over (E+N)*H
    if (i >= ETOT * NH) return;
    int e = i >> 3, hh = i & 7, s, d;
    edge_sd(ei, e, s, d);
    float v = edge_e(as, ad, s, d, hh);
    atomicMax(emax + d * NH + hh, f2ord(v));
}

// ---------- pass 2: segment sum of exp ----------
__global__ __launch_bounds__(256) void k_denom(const int* __restrict__ ei,
                                               const float* __restrict__ as,
                                               const float* __restrict__ ad,
                                               const unsigned* __restrict__ emax,
                                               float* __restrict__ denom) {
    int i = blockIdx.x * blockDim.x + threadIdx.x;
    if (i >= ETOT * NH) return;
    int e = i >> 3, hh = i & 7, s, d;
    edge_sd(ei, e, s, d);
    float v  = edge_e(as, ad, s, d, hh);
    float m  = ord2f(emax[d * NH + hh]);
    float ex = __expf(v - m);
    atomicAdd(denom + d * NH + hh, ex);
}

// ---------- pass 3: alpha-weighted scatter of source messages ----------
__global__ __launch_bounds__(256) void k_scatter(const int* __restrict__ ei,
                                                 const float* __restrict__ as,
                                                 const float* __restrict__ ad,
                                                 const unsigned* __restrict__ emax,
                                                 const float* __restrict__ denom,
                                                 const float* __restrict__ h,
                                                 float* __restrict__ out) {
    int i = blockIdx.x * blockDim.x + threadIdx.x;
    if (i >= ETOT * NH) return;
    int e = i >> 3, hh = i & 7, s, d;
    edge_sd(ei, e, s, d);

    const float* hp = h + (long)s * HC + hh * NC;
    __builtin_prefetch(hp, 0, 0);                      // global_prefetch_b8

    float v     = edge_e(as, ad, s, d, hh);
    float m     = ord2f(emax[d * NH + hh]);
    float ex    = __expf(v - m);
    float alpha = ex / (denom[d * NH + hh] + 1e-16f);

    float4 h0 = ((const float4*)hp)[0];
    float4 h1 = ((const float4*)hp)[1];
    float* op = out + (long)d * HC + hh * NC;
    atomicAdd(op + 0, alpha * h0.x);
    atomicAdd(op + 1, alpha * h0.y);
    atomicAdd(op + 2, alpha * h0.z);
    atomicAdd(op + 3, alpha * h0.w);
    atomicAdd(op + 4, alpha * h1.x);
    atomicAdd(op + 5, alpha * h1.y);
    atomicAdd(op + 6, alpha * h1.z);
    atomicAdd(op + 7, alpha * h1.w);
}

// ---------- bias + ELU ----------
__global__ __launch_bounds__(256) void k_final(float* __restrict__ out,
                                               const float* __restrict__ bias) {
    int i = blockIdx.x * blockDim.x + threadIdx.x;
    if (i >= NNODES * HC) return;
    float v = out[i] + bias[i & (HC - 1)];
    out[i] = v > 0.0f ? v : (__expf(v) - 1.0f);
}

extern "C" void kernel_launch(void* const* d_in, const int* in_sizes, int n_in,
                              void* d_out, int out_size, void* d_ws, size_t ws_size,
                              hipStream_t stream) {
    const float* x       = (const float*)d_in[0];
    const int*   ei      = (const int*)  d_in[1];   // edge_index [2, E]
    const float* W       = (const float*)d_in[2];
    const float* att_src = (const float*)d_in[3];
    const float* att_dst = (const float*)d_in[4];
    const float* bias    = (const float*)d_in[5];
    float* out = (float*)d_out;

    // workspace carve (~38.4 MB)
    char* ws = (char*)d_ws;
    float*    h      = (float*)ws;     ws += sizeof(float)    * NNODES * HC;
    float*    a_src  = (float*)ws;     ws += sizeof(float)    * NNODES * NH;
    float*    a_dst  = (float*)ws;     ws += sizeof(float)    * NNODES * NH;
    unsigned* emax   = (unsigned*)ws;  ws += sizeof(unsigned) * NNODES * NH;
    float*    denom  = (float*)ws;

    const int ntile   = NNODES / 16;                      // 6250 (exact)
    const int eh      = ETOT * NH;                        // 13.6M
    const int blk     = 256;

    k_init   <<<(NNODES * HC + blk - 1) / blk, blk, 0, stream>>>(out, denom, emax);
    k_gemm   <<<(ntile + 7) / 8,               blk, 0, stream>>>(x, W, h);
    k_att    <<<(NNODES * NH + blk - 1) / blk, blk, 0, stream>>>(h, att_src, att_dst, a_src, a_dst);
    k_emax   <<<(eh + blk - 1) / blk,          blk, 0, stream>>>(ei, a_src, a_dst, emax);
    k_denom  <<<(eh + blk - 1) / blk,          blk, 0, stream>>>(ei, a_src, a_dst, emax, denom);
    k_scatter<<<(eh + blk - 1) / blk,          blk, 0, stream>>>(ei, a_src, a_dst, emax, denom, h, out);
    k_final  <<<(NNODES * HC + blk - 1) / blk, blk, 0, stream>>>(out, bias);
}